// RecalibrationLoss_36893769072877
// MI455X (gfx1250) — compile-verified
//
#include <hip/hip_runtime.h>
#include <cstdint>

#define IOU_EPS      1e-6f
#define TILE_M       4000      // gt tile staged in LDS (64 KB boxes + 16 KB areas)
#define BLK          256       // 8 waves (wave32)
#define PRED_PER_BLK 32        // one pred box per lane

#if defined(__has_builtin)
#  if __has_builtin(__builtin_amdgcn_global_load_async_to_lds_b128)
#    define HAVE_ASYNC_B128 1
#  endif
#  if __has_builtin(__builtin_amdgcn_s_wait_asynccnt)
#    define HAVE_WAIT_ASYNC 1
#  endif
#endif

typedef int v4i_t __attribute__((ext_vector_type(4)));
typedef __attribute__((address_space(1))) v4i_t* glb_v4i_p;
typedef __attribute__((address_space(3))) v4i_t* lds_v4i_p;

__device__ __forceinline__ void async_copy_b128(const void* g, void* l) {
#if defined(HAVE_ASYNC_B128)
    // global -> LDS async DMA, 16 bytes per lane (ASYNCcnt-tracked).
    // AS3 pointer = low 32 bits of the generic shared address (ISA 10.2).
    __builtin_amdgcn_global_load_async_to_lds_b128(
        (glb_v4i_p)(uintptr_t)g,
        (lds_v4i_p)(uint32_t)(uintptr_t)l,
        /*offset=*/0, /*cpol=*/0);
#else
    *(float4*)l = *(const float4*)g;
#endif
}

__device__ __forceinline__ void wait_async_copies() {
#if defined(HAVE_ASYNC_B128)
#  if defined(HAVE_WAIT_ASYNC)
    __builtin_amdgcn_s_wait_asynccnt(0);
#  else
    asm volatile("s_wait_asynccnt 0" ::: "memory");
#  endif
#endif
}

// Pass 1: block handles 32 pred boxes (one per lane); wave w scans gt indices
// j == w (mod 8) from the LDS-staged tile. All 32 lanes of a wave read the
// same LDS address each iteration -> conflict-free broadcast.
__global__ void __launch_bounds__(BLK)
iou_bce_partials(const float* __restrict__ scores,   // N floats
                 const float* __restrict__ pred,     // N x 4
                 const float* __restrict__ gt,       // M x 4
                 float* __restrict__ partials,       // nBlocks floats
                 int N, int M)
{
    __shared__ float4 sGt[TILE_M];        // 64 000 B: gt boxes
    __shared__ float  sA2e[TILE_M];       // 16 000 B: gt area + eps
    __shared__ int    sFound[BLK];
    __shared__ float  sLoss[PRED_PER_BLK];

    const int t    = threadIdx.x;
    const int lane = t & 31;              // pred slot within block
    const int wave = t >> 5;              // 0..7, gt stride phase
    const int i    = blockIdx.x * PRED_PER_BLK + lane;
    const bool valid = (i < N);

    float px1 = 0.f, py1 = 0.f, px2 = 0.f, py2 = 0.f, a1 = 0.f;
    if (valid) {
        const float4 pb = ((const float4*)pred)[i];   // 16B aligned rows
        px1 = pb.x; py1 = pb.y; px2 = pb.z; py2 = pb.w;
        a1  = (px2 - px1) * (py2 - py1);              // pred area
    }

    int found = 0;
    for (int base = 0; base < M; base += TILE_M) {
        const int rem = M - base;
        const int tm  = rem < TILE_M ? rem : TILE_M;

        // Stage gt tile into LDS via async DMA.
        for (int idx = t; idx < tm; idx += BLK)
            async_copy_b128(gt + 4 * (size_t)(base + idx), &sGt[idx]);
        wait_async_copies();      // each wave drains its own ASYNCcnt
        __syncthreads();          // all waves see the full tile

        // Precompute per-gt area + eps once (amortized over 32 pred boxes).
        for (int idx = t; idx < tm; idx += BLK) {
            const float4 g = sGt[idx];
            sA2e[idx] = (g.z - g.x) * (g.w - g.y) + IOU_EPS;
        }
        __syncthreads();

#pragma unroll 4
        for (int j = wave; j < tm; j += 8) {
            const float4 g = sGt[j];                  // LDS broadcast read
            const float ix1 = fmaxf(px1, g.x);
            const float iy1 = fmaxf(py1, g.y);
            const float ix2 = fminf(px2, g.z);
            const float iy2 = fminf(py2, g.w);
            const float dx  = fmaxf(ix2 - ix1, 0.0f);
            const float dy  = fmaxf(iy2 - iy1, 0.0f);
            const float inter = dx * dy;
            // iou >= 0.5  <=>  3*inter >= a1 + a2 + eps   (no division)
            found |= (3.0f * inter >= a1 + sA2e[j]) ? 1 : 0;
        }
        __syncthreads();          // tile consumed before next overwrite
    }

    sFound[t] = found;
    __syncthreads();

    if (t < PRED_PER_BLK) {
        int tgt = 0;
#pragma unroll
        for (int w = 0; w < 8; ++w) tgt |= sFound[t + 32 * w];
        float loss = 0.0f;
        if (valid) {
            const float p   = scores[i];
            const float lp  = fmaxf(logf(p),     -100.0f);
            const float lq  = fmaxf(log1pf(-p),  -100.0f);
            loss = -(tgt ? lp : lq);
        }
        sLoss[t] = loss;
    }
    __syncthreads();

    if (t == 0) {                 // fixed-order sum -> deterministic
        float s = 0.0f;
#pragma unroll
        for (int k = 0; k < PRED_PER_BLK; ++k) s += sLoss[k];
        partials[blockIdx.x] = s;
    }
}

// Pass 2: deterministic reduction of block partials, scale by 1/N.
__global__ void __launch_bounds__(BLK)
final_reduce(const float* __restrict__ partials, int n, float invN,
             float* __restrict__ out)
{
    __shared__ float sBuf[BLK];
    const int t = threadIdx.x;
    float s = 0.0f;
    for (int j = t; j < n; j += BLK) s += partials[j];   // fixed stride order
    sBuf[t] = s;
    __syncthreads();
    if (t == 0) {
        float tot = 0.0f;
        for (int k = 0; k < BLK; ++k) tot += sBuf[k];    // fixed order
        out[0] = tot * invN;
    }
}

extern "C" void kernel_launch(void* const* d_in, const int* in_sizes, int n_in,
                              void* d_out, int out_size, void* d_ws, size_t ws_size,
                              hipStream_t stream)
{
    const float* scores = (const float*)d_in[0];   // (N,1) f32
    const float* pred   = (const float*)d_in[1];   // (N,4) f32
    const float* gtb    = (const float*)d_in[2];   // (M,4) f32
    const int N = in_sizes[1] / 4;
    const int M = in_sizes[2] / 4;

    float* partials = (float*)d_ws;                // nBlocks floats (~2.5 KB)
    const int nBlocks = (N + PRED_PER_BLK - 1) / PRED_PER_BLK;   // 625 for N=20000

    iou_bce_partials<<<nBlocks, BLK, 0, stream>>>(scores, pred, gtb, partials, N, M);
    final_reduce<<<1, BLK, 0, stream>>>(partials, nBlocks, 1.0f / (float)N,
                                        (float*)d_out);
}